// SpanMaskGenerator_69973607186867
// MI455X (gfx1250) — compile-verified
//
#include <hip/hip_runtime.h>

// SpanMaskGenerator for MI455X (gfx1250, wave32).
// Pure streaming-store kernel: 402 MB of write-once output, ~17 us HBM floor.
// No matrix math exists in this op -> WMMA is inapplicable; the CDNA5 wins are
// wave32 b128 coalesced stores with non-temporal hints (output is 2x L2 size)
// and fully branch-free register-resident span setup (no scratch spills).
//
// fp contract OFF: the JAX reference computes 0.15 + u*0.05 as separate
// mul+add roundings (XLA does not contract across HLO ops); letting hipcc
// fuse it into v_fma_f32 perturbs trunc(scale*131072) on ~1e-3 of samples.
#pragma clang fp contract(off)

typedef __attribute__((ext_vector_type(4))) float v4f;

#define SPAN_NUM_BLOCKS 4
#define THREADS 256
#define CHUNKS 4                                  // float4 chunks per thread
#define POS_PER_BLOCK (THREADS * 4 * CHUNKS)      // 4096 positions per block

__device__ __forceinline__ int imax(int a, int b) { return a > b ? a : b; }

// branch-free compare-exchange of (start,end) pairs keyed on start
__device__ __forceinline__ void cswap(int& sa, int& ea, int& sb, int& eb) {
  const bool sw = sb < sa;
  const int ts = sa, te = ea;
  sa = sw ? sb : sa;  ea = sw ? eb : ea;
  sb = sw ? ts : sb;  eb = sw ? te : eb;
}

__global__ __launch_bounds__(THREADS) void span_mask_kernel(
    const float* __restrict__ scales_u,
    const float* __restrict__ starts_u,
    float* __restrict__ ctx_out,   // [B,S] context mask as 0.0/1.0
    float* __restrict__ tgt_out,   // [B,S] target mask as 0.0/1.0
    float* __restrict__ pos_out,   // [B,S] padded masked positions (float)
    const int seq_len)
{
#pragma clang fp contract(off)
  const int b = blockIdx.y;
  const float fseq = (float)seq_len;

  // ---- per-batch span generation (uniform across block; branch-free) ----
  int s0, s1, s2, s3, e0, e1, e2, e3;
  {
    int ss[SPAN_NUM_BLOCKS], ee[SPAN_NUM_BLOCKS];
#pragma unroll
    for (int k = 0; k < SPAN_NUM_BLOCKS; ++k) {
      const float su = scales_u[b * SPAN_NUM_BLOCKS + k];
      const float ru = starts_u[b * SPAN_NUM_BLOCKS + k];
      // scales = 0.15 + u*0.05 : two separately-rounded ops (contract off)
      const float m = su * 0.05f;
      const float scale = m + 0.15f;
      int span = (int)(scale * fseq);              // trunc == floor (positive)
      span = span < 1 ? 1 : span;
      int maxs = seq_len - span;  maxs = maxs < 0 ? 0 : maxs;
      const int st = (int)(ru * ((float)maxs + 1.0f)); // mul(a, add) - no fma shape
      int en = st + span;  en = en > seq_len ? seq_len : en;
      ss[k] = st;  ee[k] = en;
    }
    s0 = ss[0]; s1 = ss[1]; s2 = ss[2]; s3 = ss[3];
    e0 = ee[0]; e1 = ee[1]; e2 = ee[2]; e3 = ee[3];
  }

  // sort 4 (start,end) pairs by start: network (0,1)(2,3)(0,2)(1,3)(1,2)
  cswap(s0, e0, s1, e1);
  cswap(s2, e2, s3, e3);
  cswap(s0, e0, s2, e2);
  cswap(s1, e1, s3, e3);
  cswap(s1, e1, s2, e2);

  // Running-max merge -> disjoint ascending coverage segments [lo_k, hi_k).
  // Newly covered part of span k is [max(s_k,E_{k-1}), max(e_k,E_{k-1})).
  const int lo0 = s0,            hi0 = e0;
  const int lo1 = imax(s1, hi0), hi1 = imax(e1, hi0);
  const int lo2 = imax(s2, hi1), hi2 = imax(e2, hi1);
  const int lo3 = imax(s3, hi2), hi3 = imax(e3, hi2);

  const int len0 = hi0 - lo0, len1 = hi1 - lo1;
  const int len2 = hi2 - lo2, len3 = hi3 - lo3;
  const int c1 = len0;
  const int c2 = c1 + len1;
  const int c3 = c2 + len2;
  const int total = c3 + len3;
  const int bias0 = lo0;        // cum0 == 0
  const int bias1 = lo1 - c1;
  const int bias2 = lo2 - c2;
  const int bias3 = lo3 - c3;

  const size_t row = (size_t)b * (size_t)seq_len;
  float* __restrict__ ctx = ctx_out + row;
  float* __restrict__ tgt = tgt_out + row;
  float* __restrict__ pp  = pos_out + row;

  const int block_base = blockIdx.x * POS_PER_BLOCK;
#pragma unroll
  for (int c = 0; c < CHUNKS; ++c) {
    const int p0 = block_base + c * (THREADS * 4) + (int)threadIdx.x * 4;
    if (p0 >= seq_len) continue;          // S=131072 -> never taken, safe guard
    v4f tv, cv, pv;
#pragma unroll
    for (int j = 0; j < 4; ++j) {
      const int p = p0 + j;
      // coverage test: 2 ops per segment via unsigned range trick
      const bool covered = ((unsigned)(p - lo0) < (unsigned)len0) |
                           ((unsigned)(p - lo1) < (unsigned)len1) |
                           ((unsigned)(p - lo2) < (unsigned)len2) |
                           ((unsigned)(p - lo3) < (unsigned)len3);
      tv[j] = covered ? 1.0f : 0.0f;
      cv[j] = covered ? 0.0f : 1.0f;
      const int v = (p < c1) ? (p + bias0)
                  : (p < c2) ? (p + bias1)
                  : (p < c3) ? (p + bias2)
                  :            (p + bias3);
      pv[j] = (p < total) ? (float)v : fseq;   // exact: values <= 2^17 < 2^24
    }
    // write-once, 2x-L2-sized output: non-temporal b128 stores (th:NT)
    __builtin_nontemporal_store(cv, (v4f*)(ctx + p0));
    __builtin_nontemporal_store(tv, (v4f*)(tgt + p0));
    __builtin_nontemporal_store(pv, (v4f*)(pp + p0));
  }
}

extern "C" void kernel_launch(void* const* d_in, const int* in_sizes, int n_in,
                              void* d_out, int out_size, void* d_ws, size_t ws_size,
                              hipStream_t stream) {
  const float* scales_u = (const float*)d_in[0];
  const float* starts_u = (const float*)d_in[1];

  // Derive shapes host-side (graph-capture safe; no device scalar reads):
  const int batch   = in_sizes[0] / SPAN_NUM_BLOCKS;              // 256
  const int seq_len = (int)((long long)out_size / (3LL * batch)); // 131072

  float* ctx_out = (float*)d_out;
  float* tgt_out = ctx_out + (size_t)batch * (size_t)seq_len;
  float* pos_out = tgt_out + (size_t)batch * (size_t)seq_len;

  const int gx = (seq_len + POS_PER_BLOCK - 1) / POS_PER_BLOCK;   // 32
  dim3 grid((unsigned)gx, (unsigned)batch);
  span_mask_kernel<<<grid, dim3(THREADS), 0, stream>>>(
      scales_u, starts_u, ctx_out, tgt_out, pos_out, seq_len);
}